// ResidentTransformer_81131932221764
// MI455X (gfx1250) — compile-verified
//
#include <hip/hip_runtime.h>
#include <hip/hip_bf16.h>
#include <math.h>

// ---------------------------------------------------------------------------
// ResidentTransformer decode step for MI455X (gfx1250).
// L=4, B=4, NH=16, T=2048, HD=128, H=2048, FF=8192, V=32000, S=1, fp32.
// Memory-bound GEMV workload: ~1.5GB streamed / 23.3 TB/s => ~65us floor.
// Big GEMVs use V_WMMA_F32_16X16X4_F32 (exact fp32 matrix pipe); all
// zero-reuse streams (weights, KV cache) use non-temporal loads so the
// 192MB L2 keeps only the reused activations.
// ---------------------------------------------------------------------------

typedef float v2f __attribute__((ext_vector_type(2)));
typedef float v4f __attribute__((ext_vector_type(4)));
typedef float v8f __attribute__((ext_vector_type(8)));

#define LN_EPS 1e-5f

// ---------------------------------------------------------------------------
// Positional encoding add:  X[b,i] = xin[b,i] + PE(pos, i)
// ---------------------------------------------------------------------------
__global__ __launch_bounds__(256) void pe_add_kernel(
    const float* __restrict__ xin, float* __restrict__ X,
    const int* __restrict__ pos_p, int H) {
  const int b = blockIdx.x;
  const float pos = (float)pos_p[0];
  const float neg_ln1e4_over_d = -logf(10000.0f) / (float)H;
  for (int i = threadIdx.x; i < H; i += 256) {
    const int even = i & ~1;                       // arange(0,d,2) index
    const float div = expf((float)even * neg_ln1e4_over_d);
    const float ang = pos * div;
    const float pe = (i & 1) ? cosf(ang) : sinf(ang);
    X[(size_t)b * H + i] = xin[(size_t)b * H + i] + pe;
  }
}

// ---------------------------------------------------------------------------
// LayerNorm over last dim, one block per row (B rows).
// ---------------------------------------------------------------------------
__global__ __launch_bounds__(256) void layernorm_kernel(
    const float* __restrict__ in, float* __restrict__ out,
    const float* __restrict__ w, const float* __restrict__ bs, int H) {
  __shared__ float sm[256];
  const int b = blockIdx.x;
  const int tid = threadIdx.x;
  const float* row = in + (size_t)b * H;

  float s = 0.0f;
  for (int i = tid; i < H; i += 256) s += row[i];
  sm[tid] = s; __syncthreads();
  for (int st = 128; st > 0; st >>= 1) {
    if (tid < st) sm[tid] += sm[tid + st];
    __syncthreads();
  }
  const float mu = sm[0] / (float)H;
  __syncthreads();

  float v = 0.0f;
  for (int i = tid; i < H; i += 256) { float d = row[i] - mu; v += d * d; }
  sm[tid] = v; __syncthreads();
  for (int st = 128; st > 0; st >>= 1) {
    if (tid < st) sm[tid] += sm[tid + st];
    __syncthreads();
  }
  const float var = sm[0] / (float)H;
  const float inv = rsqrtf(var + LN_EPS);

  for (int i = tid; i < H; i += 256)
    out[(size_t)b * H + i] = (row[i] - mu) * inv * w[i] + bs[i];
}

// ---------------------------------------------------------------------------
// GEMV via fp32 WMMA:  y[n, o] = sum_k x[n, k] * W[o, k]
//   W row-major [O, K] (K compile-time), O % 128 == 0, Bn <= 16 batch rows.
// Each wave owns a 16-row output tile; K loop in steps of 4 with
// V_WMMA_F32_16X16X4_F32.  Register layouts per CDNA5 ISA 7.12.2:
//   A 16x4 : lanes 0-15 -> M=0..15 {K=0,K=1}; lanes 16-31 -> M=0..15 {K=2,K=3}
//   B 4x16 : lanes 0-15 -> N=0..15 {K=0,K=1}; lanes 16-31 -> N=0..15 {K=2,K=3}
//   C/D    : lane (h*16+n) holds rows M=h*8..h*8+7 of column N=n
// Weights are streamed non-temporally (no reuse; keep them out of L2/WGP$).
// B columns N >= Bn are dead (never stored): the x-row index is clamped so
// every lane loads unconditionally -> branch-free loop, uniform EXEC, no
// per-iteration exec save/restore around the WMMA.
// ---------------------------------------------------------------------------
template <int K>
__global__ __launch_bounds__(256) void gemv_wmma_kernel(
    const float* __restrict__ W, const float* __restrict__ x,
    float* __restrict__ y, int O, int Bn) {
  const int wave = threadIdx.x >> 5;                     // 0..7
  const int lane = threadIdx.x & 31;
  const int o0   = (blockIdx.x * 8 + wave) * 16;         // tile base output row
  const int mn   = lane & 15;                            // M for A, N for B/D
  const int kk   = (lane >> 4) << 1;                     // 0 or 2 (K sub-pair)
  const int xr   = (mn < Bn) ? mn : (Bn - 1);            // clamp: dead cols OK

  const float* wrow = W + (size_t)(o0 + mn) * K + kk;
  const float* xrow = x + (size_t)xr * K + kk;

  v8f c = {0.f, 0.f, 0.f, 0.f, 0.f, 0.f, 0.f, 0.f};

#pragma unroll 8
  for (int k0 = 0; k0 < K; k0 += 4) {
    v2f a = __builtin_nontemporal_load((const v2f*)(wrow + k0));
    v2f b = *(const v2f*)(xrow + k0);                    // reused: keep cached
    c = __builtin_amdgcn_wmma_f32_16x16x4_f32(
        /*neg_a=*/false, a, /*neg_b=*/false, b,
        /*c_mod=*/(short)0, c, /*reuse_a=*/false, /*reuse_b=*/false);
  }

  if (mn < Bn) {                                         // column = batch row
    const int mbase = (lane >> 4) * 8;                   // rows 0..7 or 8..15
    float* dst = y + (size_t)mn * O + o0 + mbase;
#pragma unroll
    for (int r = 0; r < 8; ++r) dst[r] = c[r];
  }
}

// ---------------------------------------------------------------------------
// Single-token attention for one (b, head): scores over T cached keys with
// the newest K/V substituted at t = T-1 (inputs are never mutated).
// x[b, h*HD + d] += attn_out[d]   (residual add fused)
// KV cache is streamed non-temporally (537MB/call, zero reuse).
// ---------------------------------------------------------------------------
__global__ __launch_bounds__(256) void attn_kernel(
    const float* __restrict__ qkv,   // [B, 3H], H = NH*HD
    const float* __restrict__ kc,    // [B, NH, T, HD]
    const float* __restrict__ vc,    // [B, NH, T, HD]
    float* __restrict__ x,           // [B, H] residual target
    int B, int NH, int T, int HD) {
  __shared__ float sc[2048];         // T <= 2048
  __shared__ float qs[128];          // HD <= 128
  __shared__ float red[256];

  const int b = blockIdx.x / NH;
  const int h = blockIdx.x % NH;
  const int tid = threadIdx.x;
  const int H3 = 3 * NH * HD;

  const float* qn = qkv + (size_t)b * H3 + (size_t)h * 3 * HD; // new q
  const float* kn = qn + HD;                                   // new k
  const float* vn = qn + 2 * HD;                               // new v
  const float* Kb = kc + ((size_t)(b * NH + h)) * T * HD;
  const float* Vb = vc + ((size_t)(b * NH + h)) * T * HD;

  if (tid < HD) qs[tid] = qn[tid];
  __syncthreads();

  const float scale = rsqrtf((float)HD);

  // scores: thread-per-key, contiguous 512B row per key (full cachelines)
  for (int t = tid; t < T; t += 256) {
    float acc = 0.0f;
    if (t == T - 1) {
#pragma unroll 4
      for (int d = 0; d < 128; d += 4) {
        v4f kv = *(const v4f*)(kn + d);
        acc += qs[d] * kv[0] + qs[d+1] * kv[1] + qs[d+2] * kv[2] + qs[d+3] * kv[3];
      }
    } else {
      const float* kr = Kb + (size_t)t * HD;
#pragma unroll 4
      for (int d = 0; d < 128; d += 4) {
        v4f kv = __builtin_nontemporal_load((const v4f*)(kr + d));
        acc += qs[d] * kv[0] + qs[d+1] * kv[1] + qs[d+2] * kv[2] + qs[d+3] * kv[3];
      }
    }
    sc[t] = acc * scale;
  }
  __syncthreads();

  // softmax over sc[0..T)
  float m = -INFINITY;
  for (int t = tid; t < T; t += 256) m = fmaxf(m, sc[t]);
  red[tid] = m; __syncthreads();
  for (int st = 128; st > 0; st >>= 1) {
    if (tid < st) red[tid] = fmaxf(red[tid], red[tid + st]);
    __syncthreads();
  }
  m = red[0]; __syncthreads();

  float sum = 0.0f;
  for (int t = tid; t < T; t += 256) { float e = expf(sc[t] - m); sc[t] = e; sum += e; }
  red[tid] = sum; __syncthreads();
  for (int st = 128; st > 0; st >>= 1) {
    if (tid < st) red[tid] += red[tid + st];
    __syncthreads();
  }
  const float inv = 1.0f / red[0];
  __syncthreads();
  for (int t = tid; t < T; t += 256) sc[t] *= inv;
  __syncthreads();

  // attn_out[d] = sum_t p[t] * V[t,d]; split T over two half-thread-sets.
  // Loads coalesced across d lanes; V streamed non-temporally.
  const int d = tid & 127;
  const int half = tid >> 7;                 // 0 or 1
  const int t0 = half * (T / 2);
  const int t1 = t0 + (T / 2);
  const int tend = (t1 == T) ? (T - 1) : t1; // newest token handled separately
  float acc = 0.0f;
  for (int t = t0; t < tend; ++t)
    acc += sc[t] * __builtin_nontemporal_load(Vb + (size_t)t * HD + d);
  if (t1 == T) acc += sc[T - 1] * vn[d];

  red[tid] = acc; __syncthreads();
  if (half == 0)
    x[(size_t)b * NH * HD + (size_t)h * HD + d] += red[d] + red[128 + d];
}

// ---------------------------------------------------------------------------
// Exact GELU (erf form), in place.
// ---------------------------------------------------------------------------
__global__ __launch_bounds__(256) void gelu_kernel(float* __restrict__ a, int n) {
  int i = blockIdx.x * 256 + threadIdx.x;
  if (i < n) {
    float v = a[i];
    a[i] = 0.5f * v * (1.0f + erff(v * 0.70710678118654752f));
  }
}

// ---------------------------------------------------------------------------
// Residual add: x += h
// ---------------------------------------------------------------------------
__global__ __launch_bounds__(256) void add_kernel(
    float* __restrict__ x, const float* __restrict__ h, int n) {
  int i = blockIdx.x * 256 + threadIdx.x;
  if (i < n) x[i] += h[i];
}

// ---------------------------------------------------------------------------
// Row softmax over V, one block per batch row.
// ---------------------------------------------------------------------------
__global__ __launch_bounds__(256) void softmax_kernel(
    const float* __restrict__ in, float* __restrict__ out, int V) {
  __shared__ float sm[256];
  const int b = blockIdx.x;
  const int tid = threadIdx.x;
  const float* row = in + (size_t)b * V;

  float m = -INFINITY;
  for (int i = tid; i < V; i += 256) m = fmaxf(m, row[i]);
  sm[tid] = m; __syncthreads();
  for (int st = 128; st > 0; st >>= 1) {
    if (tid < st) sm[tid] = fmaxf(sm[tid], sm[tid + st]);
    __syncthreads();
  }
  m = sm[0]; __syncthreads();

  float sum = 0.0f;
  for (int i = tid; i < V; i += 256) sum += expf(row[i] - m);
  sm[tid] = sum; __syncthreads();
  for (int st = 128; st > 0; st >>= 1) {
    if (tid < st) sm[tid] += sm[tid + st];
    __syncthreads();
  }
  const float inv = 1.0f / sm[0];

  for (int i = tid; i < V; i += 256)
    out[(size_t)b * V + i] = expf(row[i] - m) * inv;
}

// ---------------------------------------------------------------------------
// Host-side orchestration
// ---------------------------------------------------------------------------
extern "C" void kernel_launch(void* const* d_in, const int* in_sizes, int n_in,
                              void* d_out, int out_size, void* d_ws, size_t ws_size,
                              hipStream_t stream) {
  (void)in_sizes; (void)n_in; (void)out_size; (void)ws_size;

  const int L = 4, B = 4, NH = 16, T = 2048, HD = 128;
  const int H = NH * HD;           // 2048
  const int H3 = 3 * H;            // 6144
  const int FF = 4 * H;            // 8192
  const int V = 32000;

  const float* x_in     = (const float*)d_in[0];
  const float* qkv_full = (const float*)d_in[1];
  const float* k_full   = (const float*)d_in[2];
  const float* v_full   = (const float*)d_in[3];
  const float* ffn1     = (const float*)d_in[4];
  const float* ffn2     = (const float*)d_in[5];
  const float* out_w    = (const float*)d_in[6];
  const float* ln_w     = (const float*)d_in[7];
  const float* ln_b     = (const float*)d_in[8];
  const int*   cur_pos  = (const int*)d_in[9];
  float* probs = (float*)d_out;

  // workspace carve-out (floats), ~840KB total
  float* X   = (float*)d_ws;           // [B, H]
  float* XN  = X   + (size_t)B * H;    // [B, H]
  float* QKV = XN  + (size_t)B * H;    // [B, 3H]
  float* H1  = QKV + (size_t)B * H3;   // [B, FF]
  float* H2  = H1  + (size_t)B * FF;   // [B, H]
  float* LG  = H2  + (size_t)B * H;    // [B, V]

  // x = x + PE(current_pos)
  pe_add_kernel<<<B, 256, 0, stream>>>(x_in, X, cur_pos, H);

  for (int i = 0; i < L; ++i) {
    // ---- attention block ----
    layernorm_kernel<<<B, 256, 0, stream>>>(X, XN, ln_w, ln_b, H);
    gemv_wmma_kernel<2048><<<H3 / 128, 256, 0, stream>>>(
        qkv_full + (size_t)i * H3 * H, XN, QKV, H3, B);
    attn_kernel<<<B * NH, 256, 0, stream>>>(
        QKV,
        k_full + (size_t)i * B * NH * T * HD,
        v_full + (size_t)i * B * NH * T * HD,
        X, B, NH, T, HD);                       // X += attn (residual fused)

    // ---- FFN block ----
    layernorm_kernel<<<B, 256, 0, stream>>>(X, XN, ln_w, ln_b, H);
    gemv_wmma_kernel<2048><<<FF / 128, 256, 0, stream>>>(
        ffn1 + (size_t)i * FF * H, XN, H1, FF, B);
    gelu_kernel<<<(B * FF + 255) / 256, 256, 0, stream>>>(H1, B * FF);
    gemv_wmma_kernel<8192><<<H / 128, 256, 0, stream>>>(
        ffn2 + (size_t)i * H * FF, H1, H2, H, B);
    add_kernel<<<(B * H + 255) / 256, 256, 0, stream>>>(X, H2, B * H);
  }

  // ---- final LN + vocab projection + softmax ----
  layernorm_kernel<<<B, 256, 0, stream>>>(X, XN, ln_w, ln_b, H);
  gemv_wmma_kernel<2048><<<V / 128, 256, 0, stream>>>(out_w, XN, LG, V, B);
  softmax_kernel<<<B, 256, 0, stream>>>(LG, probs, V);
}